// GNN_node_28432683499906
// MI455X (gfx1250) — compile-verified
//
#include <hip/hip_runtime.h>

typedef __attribute__((ext_vector_type(2))) float v2f;
typedef __attribute__((ext_vector_type(8))) float v8f;

#define EMBD 64

// ---------------------------------------------------------------------------
// GEMM: out[nrows x 64] = op(A)[nrows x 64] @ W[64 x 64] + bias
// op(A) = A                          (bnrelu_in == 0)
// op(A) = relu(A*scale[c]+shift[c])  (bnrelu_in == 1)   -- fused BN+ReLU on input
//
// One wave computes a 16x16 f32 tile via V_WMMA_F32_16X16X4_F32, K=64 in 16
// steps. 8 waves/block -> 32 rows x 64 cols per block. W and the A-tile are
// staged through LDS.
//
// Fragment layouts per CDNA5 ISA 7.12.2 (wave32):
//   A 16x4 f32 : lanes 0-15 hold M=lane, VGPR0=K0, VGPR1=K1;
//                lanes 16-31 hold M=lane-16, VGPR0=K2, VGPR1=K3
//   B 4x16 f32 : lanes 0-15 N=lane  (VGPR0=K0, VGPR1=K1);
//                lanes 16-31 N=lane-16 (VGPR0=K2, VGPR1=K3)
//   C/D 16x16  : VGPR r, lanes 0-15 -> (M=r, N=lane); lanes 16-31 -> (M=r+8)
// ---------------------------------------------------------------------------
__global__ __launch_bounds__(256) void gemm64_wmma(
    const float* __restrict__ A, const float* __restrict__ W,
    const float* __restrict__ bias,
    const float* __restrict__ in_scale, const float* __restrict__ in_shift,
    float* __restrict__ out, int nrows, int bnrelu_in)
{
    __shared__ float sW[64 * 64];   // 16 KB
    __shared__ float sA[32 * 64];   //  8 KB
    const int t       = threadIdx.x;
    const int rowbase = blockIdx.x * 32;

    // Stage the 64x64 weight matrix (coalesced, 16 floats/thread).
#pragma unroll
    for (int i = 0; i < 16; ++i)
        sW[t + i * 256] = W[t + i * 256];

    // Stage the 32x64 A tile; apply fused BN+ReLU if requested; zero-pad tail.
#pragma unroll
    for (int i = 0; i < 8; ++i) {
        int idx = t + i * 256;
        int r = idx >> 6, c = idx & 63;
        int gr = rowbase + r;
        float v = (gr < nrows) ? A[(long long)gr * EMBD + c] : 0.0f;
        if (bnrelu_in)
            v = fmaxf(v * in_scale[c] + in_shift[c], 0.0f);
        sA[idx] = v;
    }
    __syncthreads();

    const int lane = t & 31;
    const int wave = t >> 5;
    const int rt   = wave >> 2;              // row tile within block: 0..1
    const int ct   = wave & 3;               // col tile within block: 0..3
    const int half = (lane < 16) ? 0 : 2;    // K sub-offset per lane half
    const int l16  = lane & 15;
    const int mloc = rt * 16 + l16;          // local A row for this lane
    const int col  = ct * 16 + l16;          // output column for this lane

    v8f acc = {};
#pragma unroll
    for (int k = 0; k < 64; k += 4) {
        v2f a, b;
        a.x = sA[mloc * 64 + (k + half)];
        a.y = sA[mloc * 64 + (k + half + 1)];
        b.x = sW[(k + half) * 64 + col];
        b.y = sW[(k + half + 1) * 64 + col];
        acc = __builtin_amdgcn_wmma_f32_16x16x4_f32(
            /*neg_a=*/false, a, /*neg_b=*/false, b,
            /*c_mod=*/(short)0, acc, /*reuse_a=*/false, /*reuse_b=*/false);
    }

    const float bb = bias[col];
#pragma unroll
    for (int r = 0; r < 8; ++r) {
        int m = rowbase + rt * 16 + ((lane < 16) ? r : r + 8);
        if (m < nrows)
            out[(long long)m * EMBD + col] = acc[r] + bb;
    }
}

// ---------------------------------------------------------------------------
// out0 = h   (initializes the scatter accumulator with the self term)
// ---------------------------------------------------------------------------
__global__ void copy_kernel(const float* __restrict__ a, float* __restrict__ b, int n)
{
    int i = blockIdx.x * blockDim.x + threadIdx.x;
    if (i < n) b[i] = a[i];
}

// ---------------------------------------------------------------------------
// Scatter-add: out0[dst[e], c] += h[src[e], c] * ew[e]
// One thread per (edge, channel). h/out0 are 12.8 MB each -> stays in L2.
// ---------------------------------------------------------------------------
__global__ void scatter_kernel(const int* __restrict__ src, const int* __restrict__ dst,
                               const float* __restrict__ ew, const float* __restrict__ h,
                               float* __restrict__ out0, int nE)
{
    long long idx = (long long)blockIdx.x * blockDim.x + threadIdx.x;
    int e = (int)(idx >> 6);
    int c = (int)(idx & 63);
    if (e < nE) {
        // Hint the edge streams ahead (emits global_prefetch_b8; speculative).
        __builtin_prefetch(&ew[e + 2048], 0, 0);
        __builtin_prefetch(&src[e + 2048], 0, 0);
        float v = h[(long long)src[e] * EMBD + c] * ew[e];
        atomicAdd(&out0[(long long)dst[e] * EMBD + c], v);
    }
}

// ---------------------------------------------------------------------------
// Per-channel sum / sum-of-squares over the node dimension (two-pass BN).
// ---------------------------------------------------------------------------
__global__ __launch_bounds__(256) void colsum_kernel(const float* __restrict__ t,
                                                     float* __restrict__ sums,
                                                     float* __restrict__ sumsq, int n)
{
    const int c  = threadIdx.x & 63;
    const int rl = threadIdx.x >> 6;   // 0..3
    float s = 0.0f, s2 = 0.0f;
    for (int row = blockIdx.x * 4 + rl; row < n; row += gridDim.x * 4) {
        float v = t[(long long)row * EMBD + c];
        s += v; s2 += v * v;
    }
    __shared__ float sh_s[64], sh_q[64];
    if (threadIdx.x < 64) { sh_s[threadIdx.x] = 0.0f; sh_q[threadIdx.x] = 0.0f; }
    __syncthreads();
    atomicAdd(&sh_s[c], s);
    atomicAdd(&sh_q[c], s2);
    __syncthreads();
    if (threadIdx.x < 64) {
        atomicAdd(&sums[threadIdx.x],  sh_s[threadIdx.x]);
        atomicAdd(&sumsq[threadIdx.x], sh_q[threadIdx.x]);
    }
}

__global__ void zero_kernel(float* __restrict__ p, int n)
{
    int i = blockIdx.x * blockDim.x + threadIdx.x;
    if (i < n) p[i] = 0.0f;
}

// scale[c] = g[c]/sqrt(var+eps); shift[c] = be[c] - mean*scale[c]
__global__ void bn_stats_kernel(const float* __restrict__ sums, const float* __restrict__ sumsq,
                                const float* __restrict__ g, const float* __restrict__ be,
                                float* __restrict__ scale, float* __restrict__ shift, float inv_n)
{
    int c = threadIdx.x;            // 64 threads
    float m  = sums[c] * inv_n;
    float v  = sumsq[c] * inv_n - m * m;
    float sc = g[c] * rsqrtf(v + 1e-5f);
    scale[c] = sc;
    shift[c] = be[c] - m * sc;
}

// h = relu(t2*scale+shift) + h   (in place on d_out)
__global__ void finalize_kernel(const float* __restrict__ t2, const float* __restrict__ scale,
                                const float* __restrict__ shift, float* __restrict__ h, int total)
{
    int i = blockIdx.x * blockDim.x + threadIdx.x;
    if (i < total) {
        int c = i & 63;
        float v = fmaxf(t2[i] * scale[c] + shift[c], 0.0f);
        h[i] = v + h[i];
    }
}

// ---------------------------------------------------------------------------
// Launcher
// ---------------------------------------------------------------------------
extern "C" void kernel_launch(void* const* d_in, const int* in_sizes, int n_in,
                              void* d_out, int out_size, void* d_ws, size_t ws_size,
                              hipStream_t stream)
{
    const float* x      = (const float*)d_in[0];   // [N,1,64]
    const int*   ei     = (const int*)  d_in[1];   // [2,E]
    /* d_in[2] = edge_attr (unused by reference path) */
    const float* ew     = (const float*)d_in[3];   // [E]
    const float* ae_w   = (const float*)d_in[4];   // [64,64]
    const float* ae_b   = (const float*)d_in[5];   // [64]
    const float* W1     = (const float*)d_in[6];   // [L,64,64]
    const float* b1     = (const float*)d_in[7];   // [L,64]
    const float* g1     = (const float*)d_in[8];
    const float* be1    = (const float*)d_in[9];
    const float* W2     = (const float*)d_in[10];  // [L,64,64]
    const float* b2     = (const float*)d_in[11];
    const float* g_out  = (const float*)d_in[12];
    const float* be_out = (const float*)d_in[13];

    float* h = (float*)d_out;                      // [N,64], updated in place

    const int N  = in_sizes[0] / EMBD;             // 50000
    const int E  = in_sizes[3];                    // 800000
    const int L  = in_sizes[6] / (EMBD * EMBD);    // 4
    const int NF = N * EMBD;

    float* ws    = (float*)d_ws;
    float* out0  = ws;                 // NF
    float* t1    = ws + (size_t)NF;    // NF
    float* t2    = ws + (size_t)2*NF;  // NF
    float* sums  = ws + (size_t)3*NF;  // 64
    float* sumsq = sums + 64;          // 64
    float* scale = sums + 128;         // 64
    float* shift = sums + 192;         // 64

    const dim3 blk(256);
    const int gGemm = (N + 31) / 32;
    const int gElem = (NF + 255) / 256;
    const long long totEC = (long long)E * EMBD;
    const int gScat = (int)((totEC + 255) / 256);
    const float inv_n = 1.0f / (float)N;

    // Atom encoder: h = x @ ae_w + ae_b
    gemm64_wmma<<<gGemm, blk, 0, stream>>>(x, ae_w, ae_b, nullptr, nullptr, h, N, 0);

    for (int l = 0; l < L; ++l) {
        // out0 = h + scatter_add(h[src] * ew)
        copy_kernel<<<gElem, blk, 0, stream>>>(h, out0, NF);
        scatter_kernel<<<gScat, blk, 0, stream>>>(ei, ei + E, ew, h, out0, E);

        // t1 = out0 @ W1[l] + b1[l]
        gemm64_wmma<<<gGemm, blk, 0, stream>>>(out0, W1 + (size_t)l * 4096,
                                               b1 + l * 64, nullptr, nullptr, t1, N, 0);

        // BN1 batch stats -> scale/shift
        zero_kernel<<<1, 128, 0, stream>>>(sums, 128);
        colsum_kernel<<<512, blk, 0, stream>>>(t1, sums, sumsq, N);
        bn_stats_kernel<<<1, 64, 0, stream>>>(sums, sumsq, g1 + l * 64, be1 + l * 64,
                                              scale, shift, inv_n);

        // t2 = relu(BN1(t1)) @ W2[l] + b2[l]   (BN+ReLU fused into GEMM input)
        gemm64_wmma<<<gGemm, blk, 0, stream>>>(t1, W2 + (size_t)l * 4096,
                                               b2 + l * 64, scale, shift, t2, N, 1);

        // BN2 batch stats -> scale/shift
        zero_kernel<<<1, 128, 0, stream>>>(sums, 128);
        colsum_kernel<<<512, blk, 0, stream>>>(t2, sums, sumsq, N);
        bn_stats_kernel<<<1, 64, 0, stream>>>(sums, sumsq, g_out + l * 64, be_out + l * 64,
                                              scale, shift, inv_n);

        // h = relu(BN2(t2)) + h
        finalize_kernel<<<gElem, blk, 0, stream>>>(t2, scale, shift, h, NF);
    }
}